// GaussianModel_79585743995421
// MI455X (gfx1250) — compile-verified
//
#include <hip/hip_runtime.h>
#include <hip/hip_bf16.h>

// ---------------------------------------------------------------------------
// Gaussian splat field eval restructured as f32 WMMA GEMM channels (gfx1250).
//   channel 0: E   = log2(op) - 0.5*log2(e)*maha   -> w = exp2(E)
//   channel 1: r^2 = |p - g|^2
//   channel 2: T1  (SH deg-1 numerator, poly in d)
//   channel 3: T2  (SH deg-2 numerator)
//   channel 4: T3  (SH deg-3 numerator)
//   mag = max(T0 + rinv*(T1 + rinv*(T2 + rinv*T3)), 0);  rinv = rsqrt(r^2)
//   opac += w;  sig += w*mag
// B coefficients are staged per gaussian-tile into LDS (async double-buffer,
// shared by all 8 waves of the block) to cut redundant L2 traffic 8x.
// ---------------------------------------------------------------------------

#define NG   4096
#define NP   32768
#define FD   20      // point-monomial feature dim (all monomials deg <= 3)
#define NCH  5
#define GT   (NG / 16)   // gaussian tiles
#define PT   (NP / 16)   // point tiles
#define TILE_F 2048      // floats per packed B tile block (1600 used + pad)

typedef float v2f __attribute__((ext_vector_type(2)));
typedef float v8f __attribute__((ext_vector_type(8)));

#define C0f   0.28209479177387814f
#define C1f   0.4886025119029199f
#define LOG2E 1.4426950408889634f

#if __has_builtin(__builtin_amdgcn_global_load_async_to_lds_b128) && \
    __has_builtin(__builtin_amdgcn_s_wait_asynccnt)
#define USE_ASYNC_LDS 1
#else
#define USE_ASYNC_LDS 0
#endif

#if __has_builtin(__builtin_amdgcn_exp2f)
#define EXP2F(x) __builtin_amdgcn_exp2f(x)
#else
#define EXP2F(x) exp2f(x)
#endif

#if USE_ASYNC_LDS
// builtin signature (from hipcc diagnostic):
//   arg0: int __attribute__((vector_size(16))) __device__(AS1) *   (global src)
//   arg1: LDS destination pointer                                   (AS3)
//   arg2/arg3: imm offset, imm cpol
typedef int v4i_g __attribute__((vector_size(16)));
#define AS1 __attribute__((address_space(1)))
#define AS3 __attribute__((address_space(3)))
__device__ __forceinline__ AS1 v4i_g* to_gas(const void* p) {
    return (AS1 v4i_g*)((v4i_g*)(void*)p);
}
__device__ __forceinline__ AS3 v4i_g* to_lds(void* p) {
    return (AS3 v4i_g*)((v4i_g*)p);
}
#endif

// feature index order:
// 0:1  1:x 2:y 3:z  4:x2 5:y2 6:z2 7:xy 8:xz 9:yz
// 10:x3 11:y3 12:z3 13:x2y 14:x2z 15:xy2 16:y2z 17:xz2 18:yz2 19:xyz
__device__ __forceinline__ int fidx(int i, int j, int k) {
    const int e[FD][3] = {
        {0,0,0},{1,0,0},{0,1,0},{0,0,1},{2,0,0},{0,2,0},{0,0,2},{1,1,0},{1,0,1},{0,1,1},
        {3,0,0},{0,3,0},{0,0,3},{2,1,0},{2,0,1},{1,2,0},{0,2,1},{1,0,2},{0,1,2},{1,1,1}};
    for (int t = 0; t < FD; ++t)
        if (e[t][0] == i && e[t][1] == j && e[t][2] == k) return t;
    return 0;
}

__device__ __forceinline__ float ipow(float x, int n) {
    float r = 1.f;
    for (int t = 0; t < n; ++t) r *= x;
    return r;
}

__device__ __forceinline__ int binom(int n, int k) {
    const int b[4][4] = {{1,0,0,0},{1,1,0,0},{1,2,1,0},{1,3,3,1}};
    return b[n][k];
}

// accumulate K * (px-a)^i (py-b)^j (pz-c)^k expanded in the point-monomial basis
__device__ void addMono(float* c20, float K, int i, int j, int k,
                        float a, float b, float c) {
    for (int ii = 0; ii <= i; ++ii)
        for (int jj = 0; jj <= j; ++jj)
            for (int kk = 0; kk <= k; ++kk) {
                float m = (float)(binom(i, ii) * binom(j, jj) * binom(k, kk));
                m *= ipow(-a, i - ii) * ipow(-b, j - jj) * ipow(-c, k - kk);
                c20[fidx(ii, jj, kk)] += K * m;
            }
}

// packed B layout: per gaussian tile, (ch, chunk c, lane-half hi, g, pair)
//   float index within tile block = (ch*5 + c)*64 + hi*32 + gm*2 + par
//   (as v2f: (ch*5 + c)*32 + hi*16 + gm), carrying K = 4c + 2*hi + par
__device__ __forceinline__ int cbt_off(int ch, int k, int gm) {
    int c = k >> 2, rem = k & 3, hi = rem >> 1, par = rem & 1;
    return (ch * 5 + c) * 64 + hi * 32 + gm * 2 + par;
}

// -------------------------- point feature kernel ---------------------------
__global__ void gs_prep_points(const float* __restrict__ pts,
                               float* __restrict__ feats) {
    int p = blockIdx.x * blockDim.x + threadIdx.x;
    if (p >= NP) return;
    float x = pts[3 * p + 0], y = pts[3 * p + 1], z = pts[3 * p + 2];
    float f[FD];
    f[0] = 1.f;  f[1] = x;      f[2] = y;      f[3] = z;
    f[4] = x*x;  f[5] = y*y;    f[6] = z*z;
    f[7] = x*y;  f[8] = x*z;    f[9] = y*z;
    f[10] = x*x*x; f[11] = y*y*y; f[12] = z*z*z;
    f[13] = x*x*y; f[14] = x*x*z; f[15] = x*y*y;
    f[16] = y*y*z; f[17] = x*z*z; f[18] = y*z*z; f[19] = x*y*z;
    for (int k = 0; k < FD; ++k) feats[p * FD + k] = f[k];
}

// ------------------------ gaussian coefficient kernel ----------------------
__global__ void gs_prep_gauss(const float* __restrict__ xyz,
                              const float* __restrict__ sh_dc,
                              const float* __restrict__ sh_rest,
                              const float* __restrict__ scaling,
                              const float* __restrict__ rotation,
                              const float* __restrict__ opl,
                              float* __restrict__ cBt,
                              float* __restrict__ T0) {
    int g = blockIdx.x * blockDim.x + threadIdx.x;
    if (g >= NG) return;

    float a = xyz[3 * g + 0], b = xyz[3 * g + 1], c = xyz[3 * g + 2];

    float sh[16];
    sh[0] = sh_dc[g];
    for (int i = 0; i < 15; ++i) sh[1 + i] = sh_rest[15 * g + i];

    float is2[3];
    for (int d = 0; d < 3; ++d) is2[d] = __expf(-2.f * scaling[3 * g + d]);

    float qr = rotation[4 * g + 0], qx = rotation[4 * g + 1];
    float qy = rotation[4 * g + 2], qz = rotation[4 * g + 3];
    float qn = sqrtf(qr * qr + qx * qx + qy * qy + qz * qz);
    qr /= qn; qx /= qn; qy /= qn; qz /= qn;

    float R[3][3];
    R[0][0] = 1.f - 2.f * (qy * qy + qz * qz);
    R[0][1] = 2.f * (qx * qy - qr * qz);
    R[0][2] = 2.f * (qx * qz + qr * qy);
    R[1][0] = 2.f * (qx * qy + qr * qz);
    R[1][1] = 1.f - 2.f * (qx * qx + qz * qz);
    R[1][2] = 2.f * (qy * qz - qr * qx);
    R[2][0] = 2.f * (qx * qz - qr * qy);
    R[2][1] = 2.f * (qy * qz + qr * qx);
    R[2][2] = 1.f - 2.f * (qx * qx + qy * qy);

    float P[3][3];
    for (int i = 0; i < 3; ++i)
        for (int k = 0; k < 3; ++k) {
            float s = 0.f;
            for (int j = 0; j < 3; ++j) s += R[i][j] * is2[j] * R[k][j];
            P[i][k] = s;
        }

    float l  = opl[g];
    float op = 1.f / (1.f + expf(-l));

    float cM[FD], cR[FD], c1[FD], c2[FD], c3[FD];
    for (int k = 0; k < FD; ++k) { cM[k] = 0.f; cR[k] = 0.f; c1[k] = 0.f; c2[k] = 0.f; c3[k] = 0.f; }

    // r^2 channel
    addMono(cR, 1.f, 2, 0, 0, a, b, c);
    addMono(cR, 1.f, 0, 2, 0, a, b, c);
    addMono(cR, 1.f, 0, 0, 2, a, b, c);

    // maha channel
    addMono(cM, P[0][0], 2, 0, 0, a, b, c);
    addMono(cM, P[1][1], 0, 2, 0, a, b, c);
    addMono(cM, P[2][2], 0, 0, 2, a, b, c);
    addMono(cM, 2.f * P[0][1], 1, 1, 0, a, b, c);
    addMono(cM, 2.f * P[0][2], 1, 0, 1, a, b, c);
    addMono(cM, 2.f * P[1][2], 0, 1, 1, a, b, c);
    for (int k = 0; k < FD; ++k) cM[k] *= (-0.5f * LOG2E);
    cM[0] += log2f(op);

    // SH deg-1: T1(d) = C1*(s3*dx + s1*dy - s2*dz)
    addMono(c1,  C1f * sh[3], 1, 0, 0, a, b, c);
    addMono(c1,  C1f * sh[1], 0, 1, 0, a, b, c);
    addMono(c1, -C1f * sh[2], 0, 0, 1, a, b, c);

    // SH deg-2
    float A4 =  1.0925484305920792f  * sh[4];
    float A5 = -1.0925484305920792f  * sh[5];
    float A6 =  0.31539156525252005f * sh[6];
    float A7 = -1.0925484305920792f  * sh[7];
    float A8 =  0.5462742152960396f  * sh[8];
    addMono(c2, -A6 + A8, 2, 0, 0, a, b, c);   // Qxx
    addMono(c2, -A6 - A8, 0, 2, 0, a, b, c);   // Qyy
    addMono(c2, 2.f * A6, 0, 0, 2, a, b, c);   // Qzz
    addMono(c2, A4, 1, 1, 0, a, b, c);         // Qxy
    addMono(c2, A7, 1, 0, 1, a, b, c);         // Qxz
    addMono(c2, A5, 0, 1, 1, a, b, c);         // Qyz

    // SH deg-3: overall factor (-1/r^3) -> negate collected coefficients
    float B9  = -0.5900435899266435f * sh[9];
    float B10 =  2.890611442640554f  * sh[10];
    float B11 = -0.4570457994644658f * sh[11];
    float B12 =  0.3731763325901154f * sh[12];
    float B13 = -0.4570457994644658f * sh[13];
    float B14 =  1.445305721320277f  * sh[14];
    float B15 = -0.5900435899266435f * sh[15];
    addMono(c3, -(-B13 + B15),        3, 0, 0, a, b, c);  // dx^3
    addMono(c3, -(-B9  - B11),        0, 3, 0, a, b, c);  // dy^3
    addMono(c3, -(2.f * B12),         0, 0, 3, a, b, c);  // dz^3
    addMono(c3, -(3.f * B9 - B11),    2, 1, 0, a, b, c);  // dx^2 dy
    addMono(c3, -(-3.f * B12 + B14),  2, 0, 1, a, b, c);  // dx^2 dz
    addMono(c3, -(-B13 - B15),        1, 2, 0, a, b, c);  // dx dy^2
    addMono(c3, -(-3.f * B12 - B14),  0, 2, 1, a, b, c);  // dy^2 dz
    addMono(c3, -(4.f * B13 - 3.f * B15), 1, 0, 2, a, b, c); // dx dz^2
    addMono(c3, -(4.f * B11),         0, 1, 2, a, b, c);  // dy dz^2
    addMono(c3, -(B10),               1, 1, 1, a, b, c);  // dx dy dz

    float* tb = cBt + (size_t)(g >> 4) * TILE_F;
    int gm = g & 15;
    for (int k = 0; k < FD; ++k) {
        tb[cbt_off(0, k, gm)] = cM[k];
        tb[cbt_off(1, k, gm)] = cR[k];
        tb[cbt_off(2, k, gm)] = c1[k];
        tb[cbt_off(3, k, gm)] = c2[k];
        tb[cbt_off(4, k, gm)] = c3[k];
    }
    T0[g] = C0f * sh[0];
}

// ------------------------------ main kernel --------------------------------
__global__ __launch_bounds__(256) void gs_eval_wmma(
        const float* __restrict__ feats,
        const float* __restrict__ cBt,
        const float* __restrict__ T0,
        float* __restrict__ out) {
    __shared__ float sB[2][TILE_F];   // double-buffered packed B tile (2x8KB)

    const int tid  = threadIdx.x;
    const int lane = tid & 31;
    const int wv   = tid >> 5;
    const int tile = blockIdx.x * (blockDim.x >> 5) + wv;   // point tile
    const int m    = lane & 15;
    const int hi   = lane >> 4;

    // ---- A fragments: 16 points x K=20, 5 chunks of K=4 ----
    // lanes 0-15 carry K = 4c+{0,1}; lanes 16-31 carry K = 4c+{2,3}; M = lane&15
    const int row = tile * 16 + m;
    const float* fr = feats + row * FD;
    v2f afr[5];
#pragma unroll
    for (int c = 0; c < 5; ++c) {
        afr[c].x = fr[4 * c + 2 * hi + 0];
        afr[c].y = fr[4 * c + 2 * hi + 1];
    }

    float osum[8], ssum[8];
#pragma unroll
    for (int j = 0; j < 8; ++j) { osum[j] = 0.f; ssum[j] = 0.f; }

    const v8f vzero = {0.f, 0.f, 0.f, 0.f, 0.f, 0.f, 0.f, 0.f};

    // ---- prologue: stage tile 0 into sB[0] ----
#if USE_ASYNC_LDS
    {
        const float* src = cBt;
        __builtin_amdgcn_global_load_async_to_lds_b128(
            to_gas(src + tid * 4), to_lds(&sB[0][tid * 4]), 0, 0);
        __builtin_amdgcn_global_load_async_to_lds_b128(
            to_gas(src + (tid + 256) * 4), to_lds(&sB[0][(tid + 256) * 4]), 0, 0);
        __builtin_amdgcn_s_wait_asynccnt(0);
    }
#else
    {
        const float4* s4 = (const float4*)cBt;
        ((float4*)sB[0])[tid]       = s4[tid];
        ((float4*)sB[0])[tid + 256] = s4[tid + 256];
    }
#endif
    __syncthreads();

    for (int nt = 0; nt < GT; ++nt) {
        // ---- kick off copy of tile nt+1 into the other buffer ----
#if USE_ASYNC_LDS
        if (nt + 1 < GT) {
            const float* src = cBt + (size_t)(nt + 1) * TILE_F;
            float* dst = sB[(nt + 1) & 1];
            __builtin_amdgcn_global_load_async_to_lds_b128(
                to_gas(src + tid * 4), to_lds(dst + tid * 4), 0, 0);
            __builtin_amdgcn_global_load_async_to_lds_b128(
                to_gas(src + (tid + 256) * 4), to_lds(dst + (tid + 256) * 4), 0, 0);
        }
#else
        float4 pf0, pf1;
        if (nt + 1 < GT) {
            const float4* s4 = (const float4*)(cBt + (size_t)(nt + 1) * TILE_F);
            pf0 = s4[tid];
            pf1 = s4[tid + 256];
        }
#endif

        // ---- compute on buffer nt&1 ----
        const v2f* bb = (const v2f*)sB[nt & 1];
        const int boff = hi * 16 + m;
        const int gcol = nt * 16 + m;

        v8f acc[NCH];
#pragma unroll
        for (int ch = 0; ch < NCH; ++ch) {
            v8f a = vzero;
#pragma unroll
            for (int c = 0; c < 5; ++c) {
                v2f bf = bb[(ch * 5 + c) * 32 + boff];  // K = 4c+2hi+{0,1}, N = m
                a = __builtin_amdgcn_wmma_f32_16x16x4_f32(
                        false, afr[c], false, bf, (short)0, a, false, false);
            }
            acc[ch] = a;
        }
        const float t0v = T0[gcol];
#pragma unroll
        for (int j = 0; j < 8; ++j) {        // pair (M = j + 8*hi, N = m)
            float w    = EXP2F(acc[0][j]);
            float r2   = fmaxf(acc[1][j], 1e-20f);
            float rinv = rsqrtf(r2);
            float mag  = fmaf(rinv, fmaf(rinv, fmaf(rinv, acc[4][j], acc[3][j]), acc[2][j]), t0v);
            mag = fmaxf(mag, 0.f);
            osum[j] += w;
            ssum[j] = fmaf(w, mag, ssum[j]);
        }

        // ---- finish copy, hand off buffers ----
#if USE_ASYNC_LDS
        __builtin_amdgcn_s_wait_asynccnt(0);
#else
        if (nt + 1 < GT) {
            float4* d4 = (float4*)sB[(nt + 1) & 1];
            d4[tid]       = pf0;
            d4[tid + 256] = pf1;
        }
#endif
        __syncthreads();
    }

    // reduce over the 16 lanes of each half (each half sums its 16 N-columns)
#pragma unroll
    for (int j = 0; j < 8; ++j) {
        float o = osum[j], s = ssum[j];
        for (int mk = 1; mk < 16; mk <<= 1) {
            o += __shfl_xor(o, mk, 32);
            s += __shfl_xor(s, mk, 32);
        }
        if (m == j) {
            int p = tile * 16 + j + 8 * hi;  // M = j (half 0) / j+8 (half 1)
            out[p]      = o;                 // opacity sum
            out[NP + p] = s;                 // signal sum
        }
    }
}

// ------------------------------- launcher ----------------------------------
extern "C" void kernel_launch(void* const* d_in, const int* in_sizes, int n_in,
                              void* d_out, int out_size, void* d_ws, size_t ws_size,
                              hipStream_t stream) {
    const float* pts     = (const float*)d_in[0];  // network_pts (NP,3)
    const float* xyz     = (const float*)d_in[3];  // (NG,3)
    const float* sh_dc   = (const float*)d_in[4];  // (NG,1)
    const float* sh_rest = (const float*)d_in[5];  // (NG,15)
    const float* scaling = (const float*)d_in[6];  // (NG,3)
    const float* rot     = (const float*)d_in[7];  // (NG,4)
    const float* opl     = (const float*)d_in[8];  // (NG,1)
    float* out = (float*)d_out;

    float* feats = (float*)d_ws;                    // NP*FD floats
    float* cBt   = feats + (size_t)NP * FD;         // GT*TILE_F floats (packed)
    float* T0    = cBt + (size_t)GT * TILE_F;       // NG floats

    gs_prep_points<<<NP / 256, 256, 0, stream>>>(pts, feats);
    gs_prep_gauss<<<NG / 256, 256, 0, stream>>>(xyz, sh_dc, sh_rest, scaling,
                                                rot, opl, cBt, T0);
    gs_eval_wmma<<<PT / 8, 256, 0, stream>>>(feats, cBt, T0, out);
}